// REINFORCE_27934467293801
// MI455X (gfx1250) — compile-verified
//
#include <hip/hip_runtime.h>
#include <hip/hip_bf16.h>
#include <cstdint>

// ---------------------------------------------------------------------------
// REINFORCE objective + reverse discounted scan for MI455X (gfx1250).
//
// Bandwidth-bound (268 MB in, 67 MB out => ~14us roofline at 23.3 TB/s).
// CDNA5-specific path: GLOBAL_LOAD_ASYNC_TO_LDS_B128 double-buffered
// prefetch (ASYNCcnt / s_wait_asynccnt) hides HBM latency behind the serial
// scan carry chain; cum output uses non-temporal b128 stores so the 67 MB
// write stream does not pollute the 192 MB L2 that the input streams use.
// Scan parallelized via the constant-coefficient closed form:
//   scale by (gamma^8)^lane -> wave32 suffix-sum via shuffles -> unscale.
// ---------------------------------------------------------------------------

#define WAVESZ        32
#define WAVES_PER_BLK 8
#define ROWS_PER_WAVE 2
#define ELEMS         8                    // t-elements per lane per chunk
#define CHUNK         (WAVESZ * ELEMS)     // 256 t per wave per chunk
#define TMAX          2048

#define GAMMA 0.99f
#define CLIPV 5.0f

typedef float f32x4 __attribute__((ext_vector_type(4)));

// Low 32 bits of a generic pointer into LDS == wave-relative LDS byte address
// (ISA: generic->LDS mapping truncates to addr[31:0]).
__device__ __forceinline__ uint32_t lds_addr_of(const void* p) {
    return (uint32_t)(uintptr_t)p;
}

// GV-mode async copy: per-lane 16B global -> LDS. Tracked by ASYNCcnt.
__device__ __forceinline__ void async_copy_b128(uint32_t lds_byte_addr,
                                                const void* gaddr) {
    asm volatile("global_load_async_to_lds_b128 %0, %1, off"
                 :
                 : "v"(lds_byte_addr), "v"((uint64_t)(uintptr_t)gaddr)
                 : "memory");
}

template <int N>
__device__ __forceinline__ void wait_asynccnt() {
#if __has_builtin(__builtin_amdgcn_s_wait_asynccnt)
    __builtin_amdgcn_s_wait_asynccnt(N);
#else
    asm volatile("s_wait_asynccnt %0" :: "i"(N) : "memory");
#endif
}

// Fast, stable log_sigmoid: two hardware transcendentals (v_exp_f32,
// v_log_f32) instead of the libm log1p expansion. |err| ~ 1e-7.
__device__ __forceinline__ float log_sigmoid(float x) {
    const float e = __expf(-fabsf(x));
    return fminf(x, 0.0f) - __logf(1.0f + e);
}

__global__ void REINFORCE_zero_obj(float* __restrict__ p, int n) {
    int i = blockIdx.x * blockDim.x + threadIdx.x;
    if (i < n) p[i] = 0.0f;
}

__global__ __launch_bounds__(WAVES_PER_BLK * WAVESZ)
void REINFORCE_scan_kernel(const float* __restrict__ log_probs,
                           const float* __restrict__ logits,
                           const float* __restrict__ weight,
                           const float* __restrict__ baselines,
                           float* __restrict__ out_obj,   // [T], pre-zeroed
                           float* __restrict__ out_cum,   // [B,T]
                           int B, int T) {
    __shared__ float s_obj[TMAX];                                        // 8 KB
    __shared__ __align__(16) float s_stage[WAVES_PER_BLK][2][4][CHUNK];  // 64 KB

    const int tid  = threadIdx.x;
    const int wave = tid >> 5;
    const int lane = tid & 31;

    for (int i = tid; i < T; i += WAVES_PER_BLK * WAVESZ) s_obj[i] = 0.0f;
    __syncthreads();

    const float g     = GAMMA;
    float P8 = 1.0f;
    #pragma unroll
    for (int i = 0; i < ELEMS; ++i) P8 *= g;          // gamma^8
    const float scl   = powf(P8, (float)lane);        // (gamma^8)^lane >= 0.083
    const float iscl  = 1.0f / scl;                   // <= 12.1
    const float Pchk  = powf(g, (float)CHUNK);        // gamma^256 (chunk carry)

    const int nChunks = T / CHUNK;

    for (int r = 0; r < ROWS_PER_WAVE; ++r) {
        const int row = blockIdx.x * (WAVES_PER_BLK * ROWS_PER_WAVE)
                        + wave * ROWS_PER_WAVE + r;
        if (row >= B) break;
        const size_t rowOff = (size_t)row * (size_t)T;

        // ---- prologue: async-prefetch the LAST chunk into buffer 0 ----
        int buf = 0;
        {
            const size_t go = rowOff + (size_t)(nChunks - 1) * CHUNK + ELEMS * lane;
            const float* gs[4] = { logits, weight, baselines, log_probs };
            #pragma unroll
            for (int s = 0; s < 4; ++s) {
                async_copy_b128(lds_addr_of(&s_stage[wave][0][s][ELEMS * lane + 0]), gs[s] + go + 0);
                async_copy_b128(lds_addr_of(&s_stage[wave][0][s][ELEMS * lane + 4]), gs[s] + go + 4);
            }
        }

        float carry = 0.0f;   // cum at t = base + CHUNK of current chunk
        for (int c = nChunks - 1; c >= 0; --c) {
            // prefetch chunk c-1 into the other buffer, then wait for the
            // 8 in-order async loads of the current buffer to land
            if (c > 0) {
                const int nb = buf ^ 1;
                const size_t go = rowOff + (size_t)(c - 1) * CHUNK + ELEMS * lane;
                const float* gs[4] = { logits, weight, baselines, log_probs };
                #pragma unroll
                for (int s = 0; s < 4; ++s) {
                    async_copy_b128(lds_addr_of(&s_stage[wave][nb][s][ELEMS * lane + 0]), gs[s] + go + 0);
                    async_copy_b128(lds_addr_of(&s_stage[wave][nb][s][ELEMS * lane + 4]), gs[s] + go + 4);
                }
                wait_asynccnt<8>();   // oldest 8 (current buffer) complete
            } else {
                wait_asynccnt<0>();
            }

            float lg[ELEMS], wt[ELEMS], bl[ELEMS], lp[ELEMS];
            #pragma unroll
            for (int h = 0; h < 2; ++h) {
                const f32x4 vlg = *(const f32x4*)&s_stage[wave][buf][0][ELEMS * lane + 4 * h];
                const f32x4 vwt = *(const f32x4*)&s_stage[wave][buf][1][ELEMS * lane + 4 * h];
                const f32x4 vbl = *(const f32x4*)&s_stage[wave][buf][2][ELEMS * lane + 4 * h];
                const f32x4 vlp = *(const f32x4*)&s_stage[wave][buf][3][ELEMS * lane + 4 * h];
                #pragma unroll
                for (int j = 0; j < 4; ++j) {
                    lg[4 * h + j] = vlg[j];
                    wt[4 * h + j] = vwt[j];
                    bl[4 * h + j] = vbl[j];
                    lp[4 * h + j] = vlp[j];
                }
            }

            float wr[ELEMS];
            #pragma unroll
            for (int i = 0; i < ELEMS; ++i) wr[i] = wt[i] * log_sigmoid(lg[i]);

            // lane-local discounted sum: S = sum_i gamma^i * wr[i] (Horner)
            float S = wr[ELEMS - 1];
            #pragma unroll
            for (int i = ELEMS - 2; i >= 0; --i) S = wr[i] + g * S;

            // wave-level reverse scan with factor P8: scale / suffix-sum / unscale
            float W = S * scl;
            #pragma unroll
            for (int off = 1; off < WAVESZ; off <<= 1) {
                const float o = __shfl_down(W, off, WAVESZ);
                if (lane + off < WAVESZ) W += o;
            }
            const float blockcum = iscl * (W + Pchk * carry);  // cum[base+8*lane]
            float nextblock      = __shfl_down(blockcum, 1, WAVESZ);
            if (lane == WAVESZ - 1) nextblock = carry;

            float cum[ELEMS];
            cum[ELEMS - 1] = wr[ELEMS - 1] + g * nextblock;
            #pragma unroll
            for (int i = ELEMS - 2; i >= 0; --i) cum[i] = wr[i] + g * cum[i + 1];

            carry = __shfl(blockcum, 0, WAVESZ);   // cum[base] -> next chunk

            const int t0 = c * CHUNK + ELEMS * lane;

            // streaming (non-temporal) b128 stores: cum is write-once data
            #pragma unroll
            for (int h = 0; h < 2; ++h) {
                f32x4 v;
                v[0] = cum[4 * h + 0]; v[1] = cum[4 * h + 1];
                v[2] = cum[4 * h + 2]; v[3] = cum[4 * h + 3];
                __builtin_nontemporal_store(v, (f32x4*)&out_cum[rowOff + t0 + 4 * h]);
            }

            #pragma unroll
            for (int i = 0; i < ELEMS; ++i) {
                const float a = fminf(fmaxf(cum[i] - bl[i], -CLIPV), CLIPV);
                atomicAdd(&s_obj[t0 + i], a * lp[i]);
            }

            buf ^= 1;
        }
    }

    __syncthreads();
    for (int i = tid; i < T; i += WAVES_PER_BLK * WAVESZ)
        atomicAdd(&out_obj[i], s_obj[i]);
}

extern "C" void kernel_launch(void* const* d_in, const int* in_sizes, int n_in,
                              void* d_out, int out_size, void* d_ws, size_t ws_size,
                              hipStream_t stream) {
    const float* log_probs = (const float*)d_in[0];
    const float* logits    = (const float*)d_in[1];
    const float* weight    = (const float*)d_in[2];
    const float* baselines = (const float*)d_in[3];

    const int BT = in_sizes[0];          // B*T
    const int T  = out_size - BT;        // out = obj[T] ++ cum[B*T]
    const int B  = BT / T;

    float* out_obj = (float*)d_out;
    float* out_cum = (float*)d_out + T;

    REINFORCE_zero_obj<<<(T + 255) / 256, 256, 0, stream>>>(out_obj, T);

    const int rowsPerBlock = WAVES_PER_BLK * ROWS_PER_WAVE;   // 16
    const int grid = (B + rowsPerBlock - 1) / rowsPerBlock;   // 512 for B=8192
    REINFORCE_scan_kernel<<<grid, WAVES_PER_BLK * WAVESZ, 0, stream>>>(
        log_probs, logits, weight, baselines, out_obj, out_cum, B, T);
}